// DifferenceOfGaussians_69647189672061
// MI455X (gfx1250) — compile-verified
//
#include <hip/hip_runtime.h>

// ---------------------------------------------------------------------------
// DifferenceOfGaussians for MI455X (gfx1250, wave32)
//  1) build_w1d      : 1D taps = column sums of the 2D Gaussian kernels
//  2) row_conv_wmma  : horizontal conv via V_WMMA_F32_16X16X4_F32 (Toeplitz B)
//  3) col_conv_wmma  : vertical  conv via V_WMMA_F32_16X16X4_F32 (Toeplitz A)
//  4) dog_pool       : fused DoG + 3x3x3 maxpool(-inf pad) + relu + soft mask
// Both conv kernels run two independent WMMA accumulator chains per wave for
// XDL-pipe ILP.
// ---------------------------------------------------------------------------

typedef __attribute__((ext_vector_type(2))) float v2f;
typedef __attribute__((ext_vector_type(8))) float v8f;

#define WSTR 160  // per-sigma padded 1D-kernel stride (floats); taps at [16, 16+ksize)

static __device__ __forceinline__ v8f wmma_f32_16x16x4(v2f a, v2f b, v8f c) {
#if defined(__has_builtin) && __has_builtin(__builtin_amdgcn_wmma_f32_16x16x4_f32)
  return __builtin_amdgcn_wmma_f32_16x16x4_f32(false, a, false, b, (short)0, c,
                                               false, false);
#else
  asm volatile("v_wmma_f32_16x16x4_f32 %0, %1, %2, %0"
               : "+v"(c)
               : "v"(a), "v"(b));
  return c;
#endif
}

// --- 1) 1D taps: w1d[s][16+t] = sum_u W2d[s][u][t]  (outer(g,g) -> col sum = g)
__global__ void build_w1d(const float* __restrict__ w2d, float* __restrict__ w1d,
                          int ksize) {
  const int s = blockIdx.x;
  const float* k2 = w2d + (size_t)s * ksize * ksize;
  for (int t = threadIdx.x; t < WSTR; t += blockDim.x) {
    float v = 0.0f;
    const int tt = t - 16;
    if (tt >= 0 && tt < ksize) {
      for (int u = 0; u < ksize; ++u) v += k2[(size_t)u * ksize + tt];
    }
    w1d[(size_t)s * WSTR + t] = v;
  }
}

// --- 2) row pass: H[s][y][x] = sum_t in[y][x+t-R] * w[s][t]
// block = 128 threads (4 waves); each wave owns one 16x16 tile; 4 tiles share
// an LDS input band of width NC*4+48 (odd LDS stride -> conflict-free).
__global__ void row_conv_wmma(const float* __restrict__ in,
                              const float* __restrict__ w1d,
                              float* __restrict__ Hbuf,
                              int W, int Himg, int R, int NC, int BSTR) {
  extern __shared__ float smem[];
  float* band = smem;               // [16][BSTR]
  float* wl = smem + 16 * BSTR;     // [WSTR]

  const int s = blockIdx.z;
  const int y0 = blockIdx.y * 16;
  const int x0 = blockIdx.x * 64;
  const int tid = threadIdx.x;

  for (int t = tid; t < WSTR; t += 128) wl[t] = w1d[(size_t)s * WSTR + t];

  // band fill: row-outer / col-inner -> coalesced 512B bursts, no int division
  const int BandW = NC * 4 + 48;
  const float* inb = in + (size_t)y0 * W + (x0 - R);
  for (int r = 0; r < 16; ++r) {
    for (int c = tid; c < BandW; c += 128) {
      const int gx = x0 - R + c;
      band[r * BSTR + c] =
          (gx >= 0 && gx < W) ? inb[(size_t)r * W + c] : 0.0f;
    }
  }
  __syncthreads();

  const int lane = tid & 31;
  const int wv = tid >> 5;          // wave -> x sub-tile
  const int half = lane >> 4;       // 0: K pair {0,1}, 1: K pair {2,3}
  const int l16 = lane & 15;        // A: M index / B: N index

  const float* arow = band + l16 * BSTR + 16 * wv + 2 * half;
  const float* wb = wl + 16 + 2 * half - l16;

  // two independent accumulator chains for XDL ILP
  v8f acc0 = {0.f, 0.f, 0.f, 0.f, 0.f, 0.f, 0.f, 0.f};
  v8f acc1 = {0.f, 0.f, 0.f, 0.f, 0.f, 0.f, 0.f, 0.f};
  int c = 0;
  for (; c + 1 < NC; c += 2) {
    v2f a0, b0, a1, b1;
    a0.x = arow[4 * c];
    a0.y = arow[4 * c + 1];
    b0.x = wb[4 * c];
    b0.y = wb[4 * c + 1];
    a1.x = arow[4 * c + 4];
    a1.y = arow[4 * c + 5];
    b1.x = wb[4 * c + 4];
    b1.y = wb[4 * c + 5];
    acc0 = wmma_f32_16x16x4(a0, b0, acc0);
    acc1 = wmma_f32_16x16x4(a1, b1, acc1);
  }
  if (c < NC) {
    v2f a0, b0;
    a0.x = arow[4 * c];
    a0.y = arow[4 * c + 1];
    b0.x = wb[4 * c];
    b0.y = wb[4 * c + 1];
    acc0 = wmma_f32_16x16x4(a0, b0, acc0);
  }

  float* outp = Hbuf + (size_t)s * Himg * W + (size_t)y0 * W + x0 + 16 * wv + l16;
#pragma unroll
  for (int j = 0; j < 8; ++j)       // D: vgpr j -> M = j + 8*half, N = l16
    outp[(size_t)(j + 8 * half) * W] = acc0[j] + acc1[j];
}

// --- 3) col pass: G[s][y][x] = sum_t H[s][y+t-R][x] * w[s][t]
// block = 256 threads (8 waves, 8 adjacent x-tiles, same sigma). B rows are
// contiguous 16-float row segments of H -> coalesced global loads, no staging.
__global__ void col_conv_wmma(const float* __restrict__ Hbuf,
                              const float* __restrict__ w1d,
                              float* __restrict__ Gbuf,
                              int W, int Himg, int R, int NC) {
  __shared__ float wl[WSTR];
  const int s = blockIdx.z;
  const int y0 = blockIdx.y * 16;
  const int tid = threadIdx.x;
  for (int t = tid; t < WSTR; t += 256) wl[t] = w1d[(size_t)s * WSTR + t];
  __syncthreads();

  const int lane = tid & 31;
  const int wv = tid >> 5;
  const int x0 = (blockIdx.x * 8 + wv) * 16;
  const int half = lane >> 4;
  const int l16 = lane & 15;

  const float* plane = Hbuf + (size_t)s * Himg * W + x0 + l16;
  const float* wa = wl + 16 + 2 * half - l16;   // A[m][k] = w_at(k - m)

  v8f acc0 = {0.f, 0.f, 0.f, 0.f, 0.f, 0.f, 0.f, 0.f};
  v8f acc1 = {0.f, 0.f, 0.f, 0.f, 0.f, 0.f, 0.f, 0.f};
  int c = 0;
  for (; c + 1 < NC; c += 2) {
    const int kb0 = 4 * c + 2 * half;
    const int kb1 = kb0 + 4;
    v2f a0, b0, a1, b1;
    a0.x = wa[4 * c];
    a0.y = wa[4 * c + 1];
    a1.x = wa[4 * c + 4];
    a1.y = wa[4 * c + 5];
    const int r00 = y0 - R + kb0;               // B[k][n] = H[y0-R+k][x0+n]
    const int r10 = y0 - R + kb1;
    b0.x = (r00 >= 0 && r00 < Himg) ? plane[(size_t)r00 * W] : 0.0f;
    b0.y = (r00 + 1 >= 0 && r00 + 1 < Himg) ? plane[(size_t)(r00 + 1) * W] : 0.0f;
    b1.x = (r10 >= 0 && r10 < Himg) ? plane[(size_t)r10 * W] : 0.0f;
    b1.y = (r10 + 1 >= 0 && r10 + 1 < Himg) ? plane[(size_t)(r10 + 1) * W] : 0.0f;
    acc0 = wmma_f32_16x16x4(a0, b0, acc0);
    acc1 = wmma_f32_16x16x4(a1, b1, acc1);
  }
  if (c < NC) {
    const int kb = 4 * c + 2 * half;
    v2f a0, b0;
    a0.x = wa[4 * c];
    a0.y = wa[4 * c + 1];
    const int r0 = y0 - R + kb;
    b0.x = (r0 >= 0 && r0 < Himg) ? plane[(size_t)r0 * W] : 0.0f;
    b0.y = (r0 + 1 >= 0 && r0 + 1 < Himg) ? plane[(size_t)(r0 + 1) * W] : 0.0f;
    acc0 = wmma_f32_16x16x4(a0, b0, acc0);
  }

  float* outp = Gbuf + (size_t)s * Himg * W + (size_t)y0 * W + x0 + l16;
#pragma unroll
  for (int j = 0; j < 8; ++j)
    outp[(size_t)(j + 8 * half) * W] = acc0[j] + acc1[j];
}

// --- 4) fused DoG + 3x3x3 max-pool (pad -inf) + relu + soft mask
// dog computed on the fly from G; 3x18x18 LDS tile per (scale, 16x16) block.
__global__ void dog_pool(const float* __restrict__ Gbuf,
                         const float* __restrict__ sigmas,
                         const float* __restrict__ thrp,
                         float* __restrict__ out,
                         int W, int Himg, int S1) {  // S1 = #scales-1 = 50
  __shared__ float dg[3][18][18];
  const int i = blockIdx.z;
  const int y0 = blockIdx.y * 16;
  const int x0 = blockIdx.x * 16;
  const int tid = threadIdx.y * 16 + threadIdx.x;
  const float NEG = -__builtin_inff();
  const size_t HW = (size_t)Himg * W;

  for (int idx = tid; idx < 3 * 18 * 18; idx += 256) {
    const int dz = idx / 324;
    const int rem = idx - dz * 324;
    const int dy = rem / 18;
    const int dx = rem - dy * 18;
    const int j = i + dz - 1;
    const int gy = y0 + dy - 1;
    const int gx = x0 + dx - 1;
    float v = NEG;
    if (j >= 0 && j < S1 && gy >= 0 && gy < Himg && gx >= 0 && gx < W) {
      const size_t off = (size_t)gy * W + gx;
      v = (Gbuf[(size_t)j * HW + off] - Gbuf[(size_t)(j + 1) * HW + off]) *
          sigmas[j];
    }
    dg[dz][dy][dx] = v;
  }
  __syncthreads();

  const int ty = threadIdx.y, tx = threadIdx.x;
  float m = NEG;
#pragma unroll
  for (int dz = 0; dz < 3; ++dz)
#pragma unroll
    for (int dy = 0; dy < 3; ++dy)
#pragma unroll
      for (int dx = 0; dx < 3; ++dx) m = fmaxf(m, dg[dz][ty + dy][tx + dx]);

  const float d = dg[1][ty + 1][tx + 1];
  const float thr = *thrp;                       // already -threshold
  const float lm = fmaxf(m + thr, 0.0f);         // relu(pooled + thr)
  const float sm = 1.0f - lm + d + thr;          // 1 - (lm - (dog + thr))
  const size_t o = (size_t)i * HW + (size_t)(y0 + ty) * W + (x0 + tx);
  out[o] = lm;
  out[(size_t)S1 * HW + o] = sm;
}

extern "C" void kernel_launch(void* const* d_in, const int* in_sizes, int n_in,
                              void* d_out, int out_size, void* d_ws,
                              size_t ws_size, hipStream_t stream) {
  const float* x = (const float*)d_in[0];      // [1,1,H,W]
  const float* w2d = (const float*)d_in[1];    // [S,1,ksize,ksize]
  const float* sig = (const float*)d_in[2];    // [S]
  const float* thr = (const float*)d_in[3];    // scalar (= -threshold)

  const int S = in_sizes[2];                   // 51
  int W = 1;
  while (W * W < in_sizes[0]) W <<= 1;         // 1024 (power of two)
  const int Himg = W;
  int ksize = 1;
  const int k2 = in_sizes[1] / S;
  while (ksize * ksize < k2) ++ksize;          // full padded kernel size
  const int R = (ksize - 1) / 2;               // max radius

  const int Ktot = ((16 + 2 * R + 3) / 4) * 4; // GEMM K, multiple of 4
  const int NC = Ktot / 4;                     // 16x16x4 WMMA chunks
  const int BandW = Ktot + 48;                 // band width for 4 x-tiles
  const int BSTR = BandW | 1;                  // odd LDS stride, no conflicts

  float* ws = (float*)d_ws;
  float* w1d = ws;                                       // S*WSTR floats
  float* Hbuf = ws + 8192;                               // S*H*W floats
  float* Gbuf = Hbuf + (size_t)S * Himg * W;             // S*H*W floats

  build_w1d<<<S, WSTR, 0, stream>>>(w2d, w1d, ksize);

  dim3 g1(W / 64, Himg / 16, S);
  const size_t smem1 = (size_t)(16 * BSTR + WSTR) * sizeof(float);
  row_conv_wmma<<<g1, 128, smem1, stream>>>(x, w1d, Hbuf, W, Himg, R, NC, BSTR);

  dim3 g2(W / 128, Himg / 16, S);
  col_conv_wmma<<<g2, 256, 0, stream>>>(Hbuf, w1d, Gbuf, W, Himg, R, NC);

  dim3 g3(W / 16, Himg / 16, S - 1);
  dog_pool<<<g3, dim3(16, 16), 0, stream>>>(Gbuf, sig, thr, (float*)d_out, W,
                                            Himg, S - 1);
}